// MultiHeadAttentionQuantum_65481071410798
// MI455X (gfx1250) — compile-verified
//
#include <hip/hip_runtime.h>
#include <stdint.h>

// out[e] = cos(x[e] + phase[head(e)]),  phase[h] = sum_j theta[h*8+j]
// Shapes (reference): B=64, T=8192, H=8, HD=8 -> n = 33,554,432 f32.
// Memory-bound: ~268 MB streamed @ 23.3 TB/s -> ~11.5 us floor. No matmul
// structure -> WMMA inapplicable; optimize the data-movement path instead:
// gfx1250 async global->LDS double-buffer pipeline + NT cache policy both
// directions (stream > 192 MB L2, touched exactly once).

typedef float float4v __attribute__((ext_vector_type(4)));
typedef int v4i __attribute__((ext_vector_type(4)));

// Exact parameter types of __builtin_amdgcn_global_load_async_to_lds_b128
// (clang prints AS1 as '__device__', AS3 as '__shared__').
typedef __attribute__((address_space(1))) v4i* as1_v4i_p;
typedef __attribute__((address_space(3))) v4i* as3_v4i_p;

#define TPB 256
#define INV_2PI 0.15915494309189535f

// gfx12-family CPol: TH in bits [2:0]; TH_LOAD_NT == 1.
#define CPOL_LOAD_NT 1

__device__ __forceinline__ float qcos(float a) {
  // v_cos_f32 computes cos(2*pi*src); hw uses the fractional part (free range
  // reduction). Pre-scale by 1/(2*pi).
  return __builtin_amdgcn_cosf(a * INV_2PI);
}

#if defined(__HIP_DEVICE_COMPILE__) && \
    __has_builtin(__builtin_amdgcn_global_load_async_to_lds_b128)
#define HAVE_ASYNC_LDS 1
#else
#define HAVE_ASYNC_LDS 0
#endif

#if HAVE_ASYNC_LDS
__device__ __forceinline__ void async_copy_b128(const float4v* g, float4v* l) {
  // GLOBAL_LOAD_ASYNC_TO_LDS_B128: per-lane 16B global -> LDS, ASYNCcnt-tracked.
  __builtin_amdgcn_global_load_async_to_lds_b128(
      (as1_v4i_p)g, (as3_v4i_p)l, /*offset=*/0, /*cpol=*/CPOL_LOAD_NT);
}
#endif

template <int N>
__device__ __forceinline__ void wait_asynccnt() {
#if defined(__HIP_DEVICE_COMPILE__)
#if __has_builtin(__builtin_amdgcn_s_wait_asynccnt)
  __builtin_amdgcn_s_wait_asynccnt(N);
#else
  asm volatile("s_wait_asynccnt %0" ::"n"(N) : "memory");
#endif
#endif
}

__device__ __forceinline__ void wait_dscnt0() {
#if defined(__HIP_DEVICE_COMPILE__)
  // Ensure our ds_load_b128 reads have retired into VGPRs before the async
  // engine is allowed to overwrite that LDS buffer (async LDS writes are NOT
  // ordered against DS-indexed reads).
  asm volatile("s_wait_dscnt 0" ::: "memory");
#endif
}

__device__ __forceinline__ float4v cos4(float4v v, float ph) {
  float4v r;
  r.x = qcos(v.x + ph);
  r.y = qcos(v.y + ph);
  r.z = qcos(v.z + ph);
  r.w = qcos(v.w + ph);
  return r;
}

__global__ __launch_bounds__(TPB) void quantum_head_cos_kernel(
    const float* __restrict__ x, const float* __restrict__ theta,
    float* __restrict__ out, const long long n4, const long long n) {
  const long long tid = (long long)blockIdx.x * TPB + threadIdx.x;
  const long long stride = (long long)gridDim.x * TPB;  // multiple of 16 vecs

  // float4 vector v covers elements 4v..4v+3 (same head, since HD=8 and
  // vectors are 4-aligned): h = (v>>1) & 7. stride % 16 == 0 => h is
  // loop-invariant per thread across the grid-stride loop.
  const int h = (int)((tid >> 1) & 7);
  float ph = 0.f;
#pragma unroll
  for (int j = 0; j < 8; ++j) ph += theta[h * 8 + j];

  const float4v* xv = (const float4v*)x;
  float4v* ov = (float4v*)out;

#if HAVE_ASYNC_LDS
  // Double-buffered async global->LDS pipeline. Each lane stages its own 16B
  // slot; no cross-lane sharing, so per-wave ASYNCcnt waits suffice (no
  // barriers). 2 x 4KB LDS per block.
  __shared__ float4v buf[2][TPB];
  long long i = tid;
  if (i < n4) {
    async_copy_b128(&xv[i], &buf[0][threadIdx.x]);
    int cur = 0;
    long long inext = i + stride;
    while (inext < n4) {
      async_copy_b128(&xv[inext], &buf[cur ^ 1][threadIdx.x]);  // prefetch k+1
      wait_asynccnt<1>();  // tile k resident in LDS (k+1 still in flight)
      float4v v = buf[cur][threadIdx.x];
      __builtin_nontemporal_store(cos4(v, ph), &ov[i]);
      wait_dscnt0();  // LDS reads retired; buffer may be overwritten next iter
      cur ^= 1;
      i = inext;
      inext += stride;
    }
    wait_asynccnt<0>();
    float4v v = buf[cur][threadIdx.x];
    __builtin_nontemporal_store(cos4(v, ph), &ov[i]);
  }
#else
  // Fallback: direct 128-bit non-temporal streaming.
  for (long long i = tid; i < n4; i += stride) {
    float4v v = __builtin_nontemporal_load(&xv[i]);
    __builtin_nontemporal_store(cos4(v, ph), &ov[i]);
  }
#endif

  // Scalar tail for n % 4 != 0 (not hit for the reference shapes).
  const long long base = n4 * 4;
  const long long e = base + tid;
  if (e < n) {
    const int hh = (int)((e >> 3) & 7);
    float p2 = 0.f;
#pragma unroll
    for (int j = 0; j < 8; ++j) p2 += theta[hh * 8 + j];
    out[e] = qcos(x[e] + p2);
  }
}

extern "C" void kernel_launch(void* const* d_in, const int* in_sizes, int n_in,
                              void* d_out, int out_size, void* d_ws,
                              size_t ws_size, hipStream_t stream) {
  const float* x = (const float*)d_in[0];      // [B,T,H*HD] f32
  const float* theta = (const float*)d_in[1];  // [H,HD] f32
  float* out = (float*)d_out;                  // [B,T,H*HD] f32

  const long long n = (long long)in_sizes[0];  // 33,554,432
  const long long n4 = n >> 2;

  // 2048 blocks * 8 waves = 16K waves: saturates HBM; 16 float4 per thread.
  const int grid = 2048;
  hipLaunchKernelGGL(quantum_head_cos_kernel, dim3(grid), dim3(TPB), 0, stream,
                     x, theta, out, n4, n);
}